// DecoderNet_26843545600188
// MI455X (gfx1250) — compile-verified
//
#include <hip/hip_runtime.h>

// Problem constants (compile-time; harness also passes H, W as scalar inputs
// which we ignore since they are fixed in the reference).
#define B_ 4
#define T_ 8
#define G_ 64
#define H_ 120
#define W_ 160
#define PIX_ (H_ * W_)          // 19200 pixels per (b,t)
#define THREADS_ 256
#define BLOCKS_X_ (PIX_ / THREADS_)   // 75, exact

typedef __attribute__((ext_vector_type(4))) unsigned int v4u_;
typedef __attribute__((ext_vector_type(4))) int          v4i_;
typedef __attribute__((ext_vector_type(8))) int          v8i_;

// One thread per pixel; one block column (blockIdx.y) per (b,t) pair.
// Gaze table (64 points x float4 = 1KB) staged to LDS once per block via the
// CDNA5 Tensor Data Mover, then broadcast-read by all lanes in the argmin loop.
__global__ __launch_bounds__(THREADS_) void gaze_argmin_kernel(
    const float* __restrict__ df, float* __restrict__ out) {
  __shared__ float4 pts[G_];  // (x, y, st, w) per gaze point; pts[0].w == drop

  const int bt = blockIdx.y;                      // 0..31  (b*T + t)
  const float* src = df + (size_t)bt * (G_ * 4);  // df[b, t, 0, 0]

  // ---- Stage gaze tile into LDS (wave 0 only; TDM ignores EXEC, so keep it
  // in a wave-uniform branch to issue exactly one DMA per block). ----
  if (threadIdx.x < 32) {
#if defined(__gfx1250__) && __has_builtin(__builtin_amdgcn_tensor_load_to_lds)
    {
      // Low 32 bits of a generic LDS pointer are the LDS byte address
      // (aperture lives in the high word).
      unsigned lds_addr = (unsigned)(unsigned long long)(const void*)pts;
      unsigned long long ga = (unsigned long long)src;

      // D# group 0 (CDNA5 ISA):
      //   [1:0] count=1, [63:32] lds_addr, [120:64] global_addr, [127:126] type=2
      v4u_ g0 = { 1u,
                  lds_addr,
                  (unsigned)(ga & 0xFFFFFFFFu),
                  (unsigned)((ga >> 32) & 0x01FFFFFFu) | (2u << 30) };

      // D# group 1 (§8.4): 1 row of 256 dwords (G*4 floats), data_size=4B.
      //   [17:16] data_size=2 (4B)         -> w0 = 0x00020000
      //   [79:48] tensor_dim0   = 256      -> w1[31:16]
      //   [111:80] tensor_dim1  = 1        -> w2[31:16]
      //   [127:112] tile_dim0   = 256      -> w3[31:16]
      //   [143:128] tile_dim1   = 1        -> w4[15:0]
      //   [207:160] tensor_dim0_stride=256 -> w5
      //   [255:208] tensor_dim1_stride=256 -> w6[31:16] (unused but sane)
      v8i_ g1 = { (int)0x00020000u,
                  (int)(256u << 16),
                  (int)(1u << 16),
                  (int)(256u << 16),
                  1,
                  256,
                  (int)(256u << 16),
                  0 };
      v4i_ gz = { 0, 0, 0, 0 };   // groups 2/3 unused (<=2D tensor)
#if __clang_major__ >= 23
      v8i_ gz8 = { 0, 0, 0, 0, 0, 0, 0, 0 };
      __builtin_amdgcn_tensor_load_to_lds(g0, g1, gz, gz, gz8, 0);
#else
      __builtin_amdgcn_tensor_load_to_lds(g0, g1, gz, gz, 0);
#endif
#if __has_builtin(__builtin_amdgcn_s_wait_tensorcnt)
      __builtin_amdgcn_s_wait_tensorcnt(0);
#else
      asm volatile("s_wait_tensorcnt 0x0" ::: "memory");
#endif
    }
#endif
    // Belt-and-suspenders: overwrite with plain b128 loads of the identical
    // data AFTER the tensorcnt wait. Guarantees LDS correctness independent
    // of D# bit-layout details; costs 2 loads/lane once per block.
    const float4* s4 = (const float4*)src;
    pts[threadIdx.x]      = s4[threadIdx.x];
    pts[threadIdx.x + 32] = s4[threadIdx.x + 32];
  }
  __syncthreads();

  // ---- Per-pixel exact argmin over 64 gaze points ----
  const int pix = blockIdx.x * THREADS_ + threadIdx.x;  // 0..19199 (exact fit)
  const int h = pix / W_;
  const int w = pix - h * W_;
  // linspace(0,1,N): x = i * fl(1/(N-1)), matching jnp.linspace's f32 step.
  const float x = (float)w * (1.0f / (float)(W_ - 1));
  const float y = (float)h * (1.0f / (float)(H_ - 1));

  float bestd = __builtin_inff();
  int bestg = 0;
  // Compare on sqrt'ed distances with strict '<' to reproduce jnp.argmin's
  // first-min-index tie rule on dist (reference argmins AFTER sqrt).
#pragma unroll 8
  for (int g = 0; g < G_; ++g) {
    float4 p = pts[g];                 // uniform address -> LDS broadcast
    float dx = x - p.x;
    float dy = y - p.y;
    float d = sqrtf(dx * dx + dy * dy);
    if (d < bestd) { bestd = d; bestg = g; }
  }

  const float4 pb = pts[bestg];
  const float dx = x - pb.x;
  const float dy = y - pb.y;
  const float st = pb.z;
  const float drop = pts[0].w;         // driver_facing[b, t, 0, 3]
  const float mask = drop * st;

  // out[b, t, c, h, w]: channel stride = H*W, (b,t) stride = 8*H*W.
  float* o = out + (size_t)bt * (8 * PIX_) + pix;
  o[0 * PIX_] = dx * mask;
  o[1 * PIX_] = dy * mask;
  o[2 * PIX_] = (dx * dx) * mask;
  o[3 * PIX_] = (dy * dy) * mask;
  o[4 * PIX_] = (dx * dy) * mask;
  o[5 * PIX_] = bestd;
  o[6 * PIX_] = st;
  o[7 * PIX_] = drop;
}

extern "C" void kernel_launch(void* const* d_in, const int* in_sizes, int n_in,
                              void* d_out, int out_size, void* d_ws, size_t ws_size,
                              hipStream_t stream) {
  (void)in_sizes; (void)n_in; (void)d_ws; (void)ws_size; (void)out_size;
  const float* df = (const float*)d_in[0];  // (B, T, G, 4) float32
  float* out = (float*)d_out;               // (B, T, 8, H, W) float32
  dim3 grid(BLOCKS_X_, B_ * T_, 1);         // 75 x 32 blocks
  dim3 block(THREADS_, 1, 1);               // 8 wave32s per block
  gaze_argmin_kernel<<<grid, block, 0, stream>>>(df, out);
}